// GCN_64252710748427
// MI455X (gfx1250) — compile-verified
//
#include <hip/hip_runtime.h>
#include <math.h>

typedef float v2f __attribute__((ext_vector_type(2)));
typedef float v8f __attribute__((ext_vector_type(8)));

#define IN_CH  128
#define HID_CH 32
#define OUT_CH 40
#define BN_EPS 1e-5f

__device__ __forceinline__ v8f wmma_f32(v2f a, v2f b, v8f c) {
  // V_WMMA_F32_16X16X4_F32 : D = A(16x4) * B(4x16) + C(16x16), exact fp32
  return __builtin_amdgcn_wmma_f32_16x16x4_f32(
      /*neg_a=*/false, a, /*neg_b=*/false, b,
      /*c_mod=*/(short)0, c, /*reuse_a=*/false, /*reuse_b=*/false);
}

// ---------------- degree / norm ----------------
__global__ __launch_bounds__(256) void k_init_deg(float* deg, int n) {
  int i = blockIdx.x * 256 + threadIdx.x;
  if (i < n) deg[i] = 1.0f;  // self loop
}

__global__ __launch_bounds__(256) void k_deg_accum(const int* __restrict__ dst,
                                                   float* deg, int e) {
  int i = blockIdx.x * 256 + threadIdx.x;
  if (i < e) atomicAdd(&deg[dst[i]], 1.0f);
}

__global__ __launch_bounds__(256) void k_dinv(float* deg, int n) {
  int i = blockIdx.x * 256 + threadIdx.x;
  if (i < n) deg[i] = rsqrtf(deg[i]);  // deg >= 1 always (self loop)
}

// ------ GEMM1: h1[n,32] = x[n,128] @ W1[128,32], 4 M-tiles / wave (WMMA) ----
__global__ __launch_bounds__(256) void k_gemm1(const float* __restrict__ x,
                                               const float* __restrict__ W1,
                                               float* __restrict__ h1, int n) {
  int wave  = blockIdx.x * (blockDim.x >> 5) + (threadIdx.x >> 5);
  int lane  = threadIdx.x & 31;
  int mBase = wave * 64;                     // 4 x 16-row tiles per wave
  if (mBase >= n) return;
  int half = lane >> 4;   // K sub-pair selector (K: 0,1 vs 2,3)
  int lid  = lane & 15;   // M (A) / N (B,C)
  bool full = (mBase + 64 <= n);             // wave-uniform

  const float* arow[4];
#pragma unroll
  for (int j = 0; j < 4; ++j) {
    int r = mBase + j * 16 + lid;
    if (!full && r >= n) r = n - 1;          // clamp (tail wave only; guarded store)
    arow[j] = x + (long long)r * IN_CH;
  }

  v8f acc0[4] = {}; v8f acc1[4] = {};
  for (int k0 = 0; k0 < IN_CH; k0 += 4) {
    int ka = k0 + half * 2;
    v2f b0; b0.x = W1[ka * HID_CH + lid];      b0.y = W1[(ka + 1) * HID_CH + lid];
    v2f b1; b1.x = W1[ka * HID_CH + 16 + lid]; b1.y = W1[(ka + 1) * HID_CH + 16 + lid];
#pragma unroll
    for (int j = 0; j < 4; ++j) {
      v2f a; a.x = arow[j][ka]; a.y = arow[j][ka + 1];
      acc0[j] = wmma_f32(a, b0, acc0[j]);
      acc1[j] = wmma_f32(a, b1, acc1[j]);
    }
  }

  int rOff = half * 8;
  if (full) {
    // branch-free store path (no per-lane exec masking)
#pragma unroll
    for (int j = 0; j < 4; ++j) {
      float* base = h1 + (long long)(mBase + j * 16 + rOff) * HID_CH + lid;
#pragma unroll
      for (int g = 0; g < 8; ++g) {
        base[g * HID_CH]      = acc0[j][g];
        base[g * HID_CH + 16] = acc1[j][g];
      }
    }
  } else {
#pragma unroll
    for (int j = 0; j < 4; ++j) {
#pragma unroll
      for (int g = 0; g < 8; ++g) {
        int r = mBase + j * 16 + g + rOff;
        if (r < n) {
          h1[(long long)r * HID_CH + lid]      = acc0[j][g];
          h1[(long long)r * HID_CH + 16 + lid] = acc1[j][g];
        }
      }
    }
  }
}

// ---------------- self-loop init + bias (layer 1), float4 ----------------
__global__ __launch_bounds__(256) void k_selfloop1(const float* __restrict__ h1,
                                                   const float* __restrict__ dinv,
                                                   const float* __restrict__ b1,
                                                   float* __restrict__ agg1, int n) {
  int tid = blockIdx.x * 256 + threadIdx.x;
  if (tid >= n * 8) return;
  int i = tid >> 3, cg = (tid & 7) * 4;
  float w = dinv[i]; w *= w;
  float4 h = *(const float4*)(h1 + (long long)i * HID_CH + cg);
  float4 b = *(const float4*)(b1 + cg);
  float4 o; o.x = h.x * w + b.x; o.y = h.y * w + b.y;
            o.z = h.z * w + b.z; o.w = h.w * w + b.w;
  *(float4*)(agg1 + (long long)i * HID_CH + cg) = o;
}

// ---------------- edge scatter (layer 1), float4 + 4 atomics ----------------
__global__ __launch_bounds__(256) void k_scatter1(const float* __restrict__ h1,
                                                  const float* __restrict__ dinv,
                                                  const int* __restrict__ src,
                                                  const int* __restrict__ dst,
                                                  float* agg1, int e) {
  int tid = blockIdx.x * 256 + threadIdx.x;
  if (tid >= e * 8) return;
  int ed = tid >> 3, cg = (tid & 7) * 4;
  int s = src[ed], d = dst[ed];
  float nrm = dinv[s] * dinv[d];
  float4 v = *(const float4*)(h1 + (long long)s * HID_CH + cg);
  float* dp = agg1 + (long long)d * HID_CH + cg;
  atomicAdd(dp + 0, v.x * nrm);
  atomicAdd(dp + 1, v.y * nrm);
  atomicAdd(dp + 2, v.z * nrm);
  atomicAdd(dp + 3, v.w * nrm);
}

// ---------------- BatchNorm stats ----------------
__global__ __launch_bounds__(64) void k_zero_stats(float* stats) {
  stats[threadIdx.x] = 0.f;
}

__global__ __launch_bounds__(256) void k_bn_stats(const float* __restrict__ agg1,
                                                  float* __restrict__ stats, int n) {
  __shared__ float s_sum[256];
  __shared__ float s_sq[256];
  int c = threadIdx.x & 31;
  int grp = threadIdx.x >> 5;  // 0..7
  float sum = 0.f, sq = 0.f;
  for (int node = blockIdx.x * 8 + grp; node < n; node += gridDim.x * 8) {
    float v = agg1[(long long)node * HID_CH + c];
    sum += v; sq += v * v;
  }
  s_sum[threadIdx.x] = sum; s_sq[threadIdx.x] = sq;
  __syncthreads();
  if (threadIdx.x < 32) {
#pragma unroll
    for (int g = 1; g < 8; ++g) { sum += s_sum[g * 32 + c]; sq += s_sq[g * 32 + c]; }
    atomicAdd(&stats[c], sum);
    atomicAdd(&stats[HID_CH + c], sq);
  }
}

// ---------------- BN apply + ReLU, float4 ----------------
__global__ __launch_bounds__(256) void k_bn_apply(float* agg1,
                                                  const float* __restrict__ stats,
                                                  const float* __restrict__ gamma,
                                                  const float* __restrict__ beta, int n) {
  int tid = blockIdx.x * 256 + threadIdx.x;
  if (tid >= n * 8) return;
  int i = tid >> 3, cg = (tid & 7) * 4;
  float invN = 1.0f / (float)n;
  float4 sm = *(const float4*)(stats + cg);
  float4 sq = *(const float4*)(stats + HID_CH + cg);
  float4 ga = *(const float4*)(gamma + cg);
  float4 be = *(const float4*)(beta + cg);
  float4 h  = *(const float4*)(agg1 + (long long)i * HID_CH + cg);
  float m, var, scl;
  float4 o;
  m = sm.x * invN; var = sq.x * invN - m * m; scl = rsqrtf(var + BN_EPS) * ga.x;
  o.x = fmaxf((h.x - m) * scl + be.x, 0.f);
  m = sm.y * invN; var = sq.y * invN - m * m; scl = rsqrtf(var + BN_EPS) * ga.y;
  o.y = fmaxf((h.y - m) * scl + be.y, 0.f);
  m = sm.z * invN; var = sq.z * invN - m * m; scl = rsqrtf(var + BN_EPS) * ga.z;
  o.z = fmaxf((h.z - m) * scl + be.z, 0.f);
  m = sm.w * invN; var = sq.w * invN - m * m; scl = rsqrtf(var + BN_EPS) * ga.w;
  o.w = fmaxf((h.w - m) * scl + be.w, 0.f);
  *(float4*)(agg1 + (long long)i * HID_CH + cg) = o;
}

// ---------------- GEMM2: h2[n,40] = hbn[n,32] @ W2[32,40] (WMMA fp32) -------
__global__ __launch_bounds__(256) void k_gemm2(const float* __restrict__ hbn,
                                               const float* __restrict__ W2,
                                               float* __restrict__ h2, int n) {
  int wave  = blockIdx.x * (blockDim.x >> 5) + (threadIdx.x >> 5);
  int lane  = threadIdx.x & 31;
  int mBase = wave * 16;
  if (mBase >= n) return;
  int half = lane >> 4;
  int lid  = lane & 15;
  bool full = (mBase + 16 <= n);             // wave-uniform
  int row  = mBase + lid;
  if (!full && row >= n) row = n - 1;
  const float* arow = hbn + (long long)row * HID_CH;
  int col2 = 32 + lid;
  bool c2ok = (col2 < OUT_CH);

  v8f c0 = {}; v8f c1 = {}; v8f c2 = {};
  for (int k0 = 0; k0 < HID_CH; k0 += 4) {
    int ka = k0 + half * 2;
    v2f a;  a.x = arow[ka];  a.y = arow[ka + 1];
    v2f b0; b0.x = W2[ka * OUT_CH + lid];      b0.y = W2[(ka + 1) * OUT_CH + lid];
    v2f b1; b1.x = W2[ka * OUT_CH + 16 + lid]; b1.y = W2[(ka + 1) * OUT_CH + 16 + lid];
    v2f b2; b2.x = c2ok ? W2[ka * OUT_CH + col2] : 0.f;
            b2.y = c2ok ? W2[(ka + 1) * OUT_CH + col2] : 0.f;
    c0 = wmma_f32(a, b0, c0);
    c1 = wmma_f32(a, b1, c1);
    c2 = wmma_f32(a, b2, c2);
  }

  int rOff = half * 8;
  if (full) {
#pragma unroll
    for (int g = 0; g < 8; ++g) {
      float* orow = h2 + (long long)(mBase + g + rOff) * OUT_CH;
      orow[lid]      = c0[g];
      orow[16 + lid] = c1[g];
      if (c2ok) orow[col2] = c2[g];
    }
  } else {
#pragma unroll
    for (int g = 0; g < 8; ++g) {
      int r = mBase + g + rOff;
      if (r < n) {
        float* orow = h2 + (long long)r * OUT_CH;
        orow[lid]      = c0[g];
        orow[16 + lid] = c1[g];
        if (c2ok) orow[col2] = c2[g];
      }
    }
  }
}

// ---------------- self-loop init + bias (layer 2), float4 ----------------
__global__ __launch_bounds__(256) void k_selfloop2(const float* __restrict__ h2,
                                                   const float* __restrict__ dinv,
                                                   const float* __restrict__ b2,
                                                   float* __restrict__ out, int n) {
  int tid = blockIdx.x * 256 + threadIdx.x;
  if (tid >= n * 10) return;
  int i = tid / 10, cg = (tid - i * 10) * 4;
  float w = dinv[i]; w *= w;
  float4 h = *(const float4*)(h2 + (long long)i * OUT_CH + cg);
  float4 b = *(const float4*)(b2 + cg);
  float4 o; o.x = h.x * w + b.x; o.y = h.y * w + b.y;
            o.z = h.z * w + b.z; o.w = h.w * w + b.w;
  *(float4*)(out + (long long)i * OUT_CH + cg) = o;
}

// ---------------- edge scatter (layer 2), float4 + 4 atomics ----------------
__global__ __launch_bounds__(256) void k_scatter2(const float* __restrict__ h2,
                                                  const float* __restrict__ dinv,
                                                  const int* __restrict__ src,
                                                  const int* __restrict__ dst,
                                                  float* out, int e) {
  int tid = blockIdx.x * 256 + threadIdx.x;
  if (tid >= e * 10) return;
  int ed = tid / 10, cg = (tid - ed * 10) * 4;
  int s = src[ed], d = dst[ed];
  float nrm = dinv[s] * dinv[d];
  float4 v = *(const float4*)(h2 + (long long)s * OUT_CH + cg);
  float* dp = out + (long long)d * OUT_CH + cg;
  atomicAdd(dp + 0, v.x * nrm);
  atomicAdd(dp + 1, v.y * nrm);
  atomicAdd(dp + 2, v.z * nrm);
  atomicAdd(dp + 3, v.w * nrm);
}

// ---------------- row-wise log_softmax over 40 classes, 1 wave32/node ------
__global__ __launch_bounds__(256) void k_logsoftmax(float* out, int n) {
  int node = blockIdx.x * 8 + (threadIdx.x >> 5);
  int lane = threadIdx.x & 31;
  if (node >= n) return;
  float* row = out + (long long)node * OUT_CH;
  float v0 = row[lane];
  float v1 = (lane < OUT_CH - 32) ? row[32 + lane] : -INFINITY;
  float m = fmaxf(v0, v1);
#pragma unroll
  for (int off = 16; off; off >>= 1) m = fmaxf(m, __shfl_xor(m, off, 32));
  float s = expf(v0 - m) + ((lane < OUT_CH - 32) ? expf(v1 - m) : 0.f);
#pragma unroll
  for (int off = 16; off; off >>= 1) s += __shfl_xor(s, off, 32);
  float lse = m + logf(s);
  row[lane] = v0 - lse;
  if (lane < OUT_CH - 32) row[32 + lane] = v1 - lse;
}

extern "C" void kernel_launch(void* const* d_in, const int* in_sizes, int n_in,
                              void* d_out, int out_size, void* d_ws, size_t ws_size,
                              hipStream_t stream) {
  const float* x     = (const float*)d_in[0];
  const int*   eidx  = (const int*)d_in[1];  // int32 (JAX x64 disabled -> int32)
  const float* W1    = (const float*)d_in[2];
  const float* b1    = (const float*)d_in[3];
  const float* gamma = (const float*)d_in[4];
  const float* beta  = (const float*)d_in[5];
  const float* W2    = (const float*)d_in[6];
  const float* b2    = (const float*)d_in[7];
  float* out = (float*)d_out;

  const int n = in_sizes[0] / IN_CH;   // 100000
  const int e = in_sizes[1] / 2;       // 1600000
  const int* src = eidx;
  const int* dst = eidx + e;

  // workspace layout (floats)
  float* ws = (float*)d_ws;
  size_t o = 0;
  float* deg   = ws + o; o += ((size_t)n + 127) & ~(size_t)127;  // becomes dinv
  float* h1    = ws + o; o += (size_t)n * HID_CH;
  float* agg1  = ws + o; o += (size_t)n * HID_CH;
  float* stats = ws + o; o += 128;
  float* h2    = ws + o; o += (size_t)n * OUT_CH;
  (void)ws_size;

  int b;
  b = (n + 255) / 256;             k_init_deg <<<b, 256, 0, stream>>>(deg, n);
  b = (e + 255) / 256;             k_deg_accum<<<b, 256, 0, stream>>>(dst, deg, e);
  b = (n + 255) / 256;             k_dinv     <<<b, 256, 0, stream>>>(deg, n);

  int g1waves  = (n + 63) / 64;                       // 64 rows per wave
  int g1blocks = (g1waves + 7) / 8;                   // 8 waves / block
  k_gemm1<<<g1blocks, 256, 0, stream>>>(x, W1, h1, n);

  b = (n * 8 + 255) / 256;         k_selfloop1<<<b, 256, 0, stream>>>(h1, deg, b1, agg1, n);
  b = (e * 8 + 255) / 256;         k_scatter1 <<<b, 256, 0, stream>>>(h1, deg, src, dst, agg1, e);

  k_zero_stats<<<1, 64, 0, stream>>>(stats);
  k_bn_stats  <<<480, 256, 0, stream>>>(agg1, stats, n);
  b = (n * 8 + 255) / 256;         k_bn_apply <<<b, 256, 0, stream>>>(agg1, stats, gamma, beta, n);

  int g2waves  = (n + 15) / 16;
  int g2blocks = (g2waves + 7) / 8;
  k_gemm2<<<g2blocks, 256, 0, stream>>>(agg1, W2, h2, n);

  b = (n * 10 + 255) / 256;        k_selfloop2<<<b, 256, 0, stream>>>(h2, deg, b2, out, n);
  b = (e * 10 + 255) / 256;        k_scatter2 <<<b, 256, 0, stream>>>(h2, deg, src, dst, out, e);

  b = (n + 7) / 8;                 k_logsoftmax<<<b, 256, 0, stream>>>(out, n);
}